// InvariantPointAttention_5411658793432
// MI455X (gfx1250) — compile-verified
//
#include <hip/hip_runtime.h>
#include <math.h>

// ---------------- problem constants ----------------
constexpr int LEN    = 512;   // sequence length
constexpr int DMODEL = 384;
constexpr int NH     = 12;
constexpr int DSK    = 16;    // scalar q/k dim
constexpr int DSV    = 16;    // scalar v dim
constexpr int DPK3   = 12;    // point q/k dims flattened (4 pts * 3)
constexpr int DPV3   = 24;    // point v dims flattened (8 pts * 3)
constexpr int DPAIR  = 128;
constexpr int NSOUT  = 576;   // H*(16+16+16)
constexpr int NPOUT  = 576;   // H*(4+4+8)*3
constexpr int DFEAT  = 2112;  // H*(16 + 24 + 8 + 128)
constexpr float EPSV         = 1e-8f;
constexpr float SCALAR_SCALE = 0.14433756729740643f;  // (3*16)^-0.5
constexpr float POINT_SCALE  = 0.13608276348795434f;  // (3*4*4.5)^-0.5
constexpr float PAIR_SCALE   = 0.57735026918962576f;  // 3^-0.5

typedef __attribute__((ext_vector_type(2))) float v2f;
typedef __attribute__((ext_vector_type(8))) float v8f;
typedef unsigned int v4u __attribute__((ext_vector_type(4)));
typedef int v4i __attribute__((ext_vector_type(4)));
typedef int v8i __attribute__((ext_vector_type(8)));

// CDNA5 fp32 WMMA: D(16x16) = A(16x4) * B(4x16) + C.
// A lane layout: lanes 0-15 hold M=lane, {K=k,K=k+1}; lanes 16-31 hold {K=k+2,K=k+3}.
// B mirrors with N in place of M.  C/D: VGPR r -> M = r + 8*(lane>=16), N = lane&15.
__device__ __forceinline__ v8f wmma4(v2f a, v2f b, v8f c) {
  return __builtin_amdgcn_wmma_f32_16x16x4_f32(false, a, false, b, (short)0, c,
                                               false, false);
}

// ---------------------------------------------------------------------------
// Tensor Data Mover: 2D tile load Global -> LDS (D# per CDNA5 ISA sec 8.3/8.4).
// data_size = 4 bytes.  tile: tile1 rows of tile0 elements, row stride stride0.
// ---------------------------------------------------------------------------
#if defined(__has_builtin)
#if __has_builtin(__builtin_amdgcn_tensor_load_to_lds) && \
    __has_builtin(__builtin_amdgcn_s_wait_tensorcnt)
#define HAVE_TDM 1
#endif
#endif

#ifdef HAVE_TDM
__device__ __forceinline__ void tdm_load_2d(unsigned long long ga, unsigned lds_off,
                                            unsigned td0, unsigned td1,
                                            unsigned tile0, unsigned tile1,
                                            unsigned long long stride0) {
  v4u g0;
  g0[0] = 1u;                                            // count=1, user mode
  g0[1] = lds_off;                                       // lds_addr [63:32]
  g0[2] = (unsigned)(ga & 0xffffffffull);                // global_addr lo
  g0[3] = (unsigned)((ga >> 32) & 0x1ffffffull) | 0x80000000u;  // addr hi | type=2
  v8i g1;
  g1[0] = (int)(2u << 16);                               // data_size = 4B
  g1[1] = (int)((td0 & 0xffffu) << 16);                  // tensor_dim0[15:0]
  g1[2] = (int)((td0 >> 16) | ((td1 & 0xffffu) << 16));  // td0 hi | td1 lo
  g1[3] = (int)((td1 >> 16) | (tile0 << 16));            // td1 hi | tile_dim0
  g1[4] = (int)(tile1 & 0xffffu);                        // tile_dim1 (tile_dim2=0)
  g1[5] = (int)(stride0 & 0xffffffffull);                // dim0_stride lo
  g1[6] = (int)((stride0 >> 32) & 0xffffull);            // dim0_stride hi
  g1[7] = 0;
  v4i gz = {0, 0, 0, 0};
#if defined(__clang_major__) && __clang_major__ >= 23
  v8i gz8 = {0, 0, 0, 0, 0, 0, 0, 0};
  __builtin_amdgcn_tensor_load_to_lds(g0, g1, gz, gz, gz8, 0);
#else
  __builtin_amdgcn_tensor_load_to_lds(g0, g1, gz, gz, 0);
#endif
}
#endif

// Stage a 2D f32 tile into LDS: TDM when available, cooperative copy otherwise.
__device__ __forceinline__ void stage_tile(const float* __restrict__ gsrc,
                                           float* __restrict__ lds,
                                           unsigned tile0, unsigned tile1,
                                           unsigned stride0 /*elements*/) {
#ifdef HAVE_TDM
  tdm_load_2d((unsigned long long)(const void*)gsrc,
              (unsigned)(unsigned long long)(const void*)lds,  // low 32b = LDS offset
              tile0, tile1, tile0, tile1, stride0);
  __builtin_amdgcn_s_wait_tensorcnt(0);
#else
  const int lane = threadIdx.x & 31;
  for (unsigned e = lane; e < tile0 * tile1; e += 32) {
    unsigned r = e / tile0, c = e % tile0;
    lds[(size_t)r * tile0 + c] = gsrc[(size_t)r * stride0 + c];
  }
  __syncthreads();
#endif
}

// ---------------------------------------------------------------------------
// Kernel 0: transpose weights W[K x N] -> WT[N x K] so all GEMM B-operands
// become contiguous b64 loads.
// ---------------------------------------------------------------------------
__global__ void k_transpose(const float* __restrict__ W, float* __restrict__ WT,
                            int K, int N) {
  int idx = blockIdx.x * blockDim.x + threadIdx.x;
  if (idx >= K * N) return;
  int k = idx / N, n = idx % N;
  WT[(size_t)n * K + k] = W[idx];
}

// ---------------------------------------------------------------------------
// Kernel 1: fp32 GEMM  Y[M x N] = X[M x K] @ WT[N x K]^T, wave per 16x16 tile.
// Both A and B inner-loop loads are contiguous global_load_b64.
// ---------------------------------------------------------------------------
__global__ __launch_bounds__(32) void k_gemmT(const float* __restrict__ X,
                                              const float* __restrict__ WT,
                                              float* __restrict__ Y, int K, int N) {
  const int lane = threadIdx.x & 31;
  const int hf = lane >> 4, lid = lane & 15;
  const int m0 = blockIdx.x * 16, n0 = blockIdx.y * 16;
  const float* xrow = X + (size_t)(m0 + lid) * K + 2 * hf;
  const float* brow = WT + (size_t)(n0 + lid) * K + 2 * hf;
  v8f acc = {};
  for (int k = 0; k < K; k += 4) {
    v2f a = *(const v2f*)(xrow + k);
    v2f b = *(const v2f*)(brow + k);
    acc = wmma4(a, b, acc);
  }
  float* y = Y + n0 + lid;
#pragma unroll
  for (int r = 0; r < 8; ++r) y[(size_t)(m0 + 8 * hf + r) * N] = acc[r];
}

// ---------------------------------------------------------------------------
// Kernel 2: scatter scalar q/k to head-major, v_s/v_p to head-major TRANSPOSED
// (N-major for b64 B loads); rotate+translate points; q_p/k_p squared norms.
// One thread per (h,i).
// ---------------------------------------------------------------------------
__global__ void k_prep(const float* __restrict__ qkvs, const float* __restrict__ qkvp,
                       const float* __restrict__ rot, const float* __restrict__ trans,
                       float* __restrict__ qs, float* __restrict__ ks,
                       float* __restrict__ vsT, float* __restrict__ qp,
                       float* __restrict__ kp, float* __restrict__ vpT,
                       float* __restrict__ qp2, float* __restrict__ kp2) {
  int idx = blockIdx.x * blockDim.x + threadIdx.x;
  if (idx >= NH * LEN) return;
  int h = idx / LEN, i = idx % LEN;
  const float* srow = qkvs + (size_t)i * NSOUT;
#pragma unroll
  for (int d = 0; d < DSK; ++d) {
    qs[((size_t)h * LEN + i) * DSK + d] = srow[h * DSK + d];
    ks[((size_t)h * LEN + i) * DSK + d] = srow[NH * DSK + h * DSK + d];
    vsT[((size_t)h * DSV + d) * LEN + i] = srow[2 * NH * DSK + h * DSK + d];
  }
  const float* prow = qkvp + (size_t)i * NPOUT + h * 48;  // (t,c), c fastest
  const float* R = rot + (size_t)i * 9;                   // R[c][d]
  const float* t = trans + (size_t)i * 3;
  float q2 = 0.f, k2 = 0.f;
#pragma unroll
  for (int tt = 0; tt < 16; ++tt) {
    float c0 = prow[tt * 3 + 0], c1 = prow[tt * 3 + 1], c2 = prow[tt * 3 + 2];
    float r0 = c0 * R[0] + c1 * R[3] + c2 * R[6] + t[0];
    float r1 = c0 * R[1] + c1 * R[4] + c2 * R[7] + t[1];
    float r2 = c0 * R[2] + c1 * R[5] + c2 * R[8] + t[2];
    if (tt < 4) {
      float* o = qp + ((size_t)h * LEN + i) * DPK3 + tt * 3;
      o[0] = r0; o[1] = r1; o[2] = r2;
      q2 += r0 * r0 + r1 * r1 + r2 * r2;
    } else if (tt < 8) {
      float* o = kp + ((size_t)h * LEN + i) * DPK3 + (tt - 4) * 3;
      o[0] = r0; o[1] = r1; o[2] = r2;
      k2 += r0 * r0 + r1 * r1 + r2 * r2;
    } else {
      int c = tt - 8;
      vpT[((size_t)h * DPV3 + c * 3 + 0) * LEN + i] = r0;
      vpT[((size_t)h * DPV3 + c * 3 + 1) * LEN + i] = r1;
      vpT[((size_t)h * DPV3 + c * 3 + 2) * LEN + i] = r2;
    }
  }
  qp2[(size_t)h * LEN + i] = q2;
  kp2[(size_t)h * LEN + i] = k2;
}

// ---------------------------------------------------------------------------
// Kernel 3: pair bias  pb[h][i][j] = (pair[i,j,:] . Wp[:,h] + bp[h]) * PAIR_SCALE
// M = 16 consecutive (i,j) rows (fixed i, contiguous 8KB) -> TDM-staged in LDS.
// N = heads padded to 16, K = 128.
// ---------------------------------------------------------------------------
__global__ __launch_bounds__(32) void k_pairbias(const float* __restrict__ pair,
                                                 const float* __restrict__ Wp,
                                                 const float* __restrict__ bp,
                                                 float* __restrict__ pb) {
  __shared__ float smem[16 * DPAIR];  // 8 KB A-panel
  const int lane = threadIdx.x & 31;
  const int hf = lane >> 4, lid = lane & 15;
  const int j0 = blockIdx.x * 16, i = blockIdx.y;
  // 16 rows x 128 floats, fully contiguous -> one 1D DMA
  stage_tile(pair + ((size_t)i * LEN + j0) * DPAIR, smem,
             16 * DPAIR, 1, 16 * DPAIR);
  const bool bvalid = lid < NH;
  const float* arow = smem + (size_t)lid * DPAIR + 2 * hf;
  v8f acc = {};
#pragma unroll 4
  for (int k = 0; k < DPAIR; k += 4) {
    v2f a = *(const v2f*)(arow + k);  // ds_load_b64
    v2f b;
    b.x = bvalid ? Wp[(k + 2 * hf) * NH + lid] : 0.f;
    b.y = bvalid ? Wp[(k + 2 * hf + 1) * NH + lid] : 0.f;
    acc = wmma4(a, b, acc);
  }
  if (lid < NH) {
    float bb = bp[lid];
    float* o = pb + (size_t)lid * LEN * LEN + (size_t)i * LEN + j0 + 8 * hf;
#pragma unroll
    for (int r = 0; r < 8; ++r) o[r] = (acc[r] + bb) * PAIR_SCALE;
  }
}

// ---------------------------------------------------------------------------
// Kernel 4: logits[h][i][j] = SCALAR_SCALE*(q_s.k_s) + gamma*(q_p.k_p)
//                            - 0.5*gamma*(|q_p|^2 + |k_p|^2) + pb
//           gamma = POINT_SCALE * softplus(pw[h]).  Wave per (h, 16i, 16j).
// ---------------------------------------------------------------------------
__global__ __launch_bounds__(32) void k_logits(
    const float* __restrict__ qs, const float* __restrict__ ks,
    const float* __restrict__ qp, const float* __restrict__ kp,
    const float* __restrict__ qp2, const float* __restrict__ kp2,
    const float* __restrict__ pwraw, const float* __restrict__ pb,
    float* __restrict__ logits) {
  const int lane = threadIdx.x & 31;
  const int hf = lane >> 4, lid = lane & 15;
  const int i0 = blockIdx.x * 16, j0 = blockIdx.y * 16, h = blockIdx.z;
  const float gamma = POINT_SCALE * log1pf(__expf(pwraw[h]));
  v8f acc = {};
  const float* qrow = qs + ((size_t)h * LEN + i0 + lid) * DSK + 2 * hf;
  const float* krow = ks + ((size_t)h * LEN + j0 + lid) * DSK + 2 * hf;
#pragma unroll
  for (int k = 0; k < DSK; k += 4) {
    v2f a = *(const v2f*)(qrow + k);
    a.x *= SCALAR_SCALE; a.y *= SCALAR_SCALE;
    v2f b = *(const v2f*)(krow + k);
    acc = wmma4(a, b, acc);
  }
  const float* qprow = qp + ((size_t)h * LEN + i0 + lid) * DPK3 + 2 * hf;
  const float* kprow = kp + ((size_t)h * LEN + j0 + lid) * DPK3 + 2 * hf;
#pragma unroll
  for (int k = 0; k < DPK3; k += 4) {
    v2f a = *(const v2f*)(qprow + k);
    a.x *= gamma; a.y *= gamma;
    v2f b = *(const v2f*)(kprow + k);
    acc = wmma4(a, b, acc);
  }
  const int j = j0 + lid;
  const float k2j = kp2[(size_t)h * LEN + j];
  const float* pbp = pb + (size_t)h * LEN * LEN + j;
  float* lp = logits + (size_t)h * LEN * LEN + j;
#pragma unroll
  for (int r = 0; r < 8; ++r) {
    int ii = i0 + 8 * hf + r;
    lp[(size_t)ii * LEN] = acc[r] + pbp[(size_t)ii * LEN] -
                           0.5f * gamma * (qp2[(size_t)h * LEN + ii] + k2j);
  }
}

// ---------------------------------------------------------------------------
// Kernel 5: in-place row softmax over j (512 per row), one wave per (h,i).
// ---------------------------------------------------------------------------
__global__ __launch_bounds__(256) void k_softmax(float* __restrict__ logits) {
  const int wave = threadIdx.x >> 5, lane = threadIdx.x & 31;
  const size_t row = (size_t)blockIdx.x * 8 + wave;  // NH*LEN rows total
  float* p = logits + row * LEN;
  float x[16];
  float m = -3.402823466e38f;
#pragma unroll
  for (int t = 0; t < 16; ++t) { x[t] = p[lane + 32 * t]; m = fmaxf(m, x[t]); }
#pragma unroll
  for (int off = 16; off >= 1; off >>= 1) m = fmaxf(m, __shfl_xor(m, off, 32));
  float s = 0.f;
#pragma unroll
  for (int t = 0; t < 16; ++t) { x[t] = __expf(x[t] - m); s += x[t]; }
#pragma unroll
  for (int off = 16; off >= 1; off >>= 1) s += __shfl_xor(s, off, 32);
  const float inv = 1.0f / s;
#pragma unroll
  for (int t = 0; t < 16; ++t) p[lane + 32 * t] = x[t] * inv;
}

// ---------------------------------------------------------------------------
// Kernel 6: res_s[h] = attn[h] @ v_s[h]  (N=16)   and
//           res_p[h] = attn[h] @ v_p[h]  (N=24, two padded tiles).  K=512.
// v_s/v_p stored N-major -> both operands are b64 loads.
// ---------------------------------------------------------------------------
__global__ __launch_bounds__(32) void k_attn_v(const float* __restrict__ attn,
                                               const float* __restrict__ vsT,
                                               const float* __restrict__ vpT,
                                               float* __restrict__ res_s,
                                               float* __restrict__ res_p) {
  const int lane = threadIdx.x & 31;
  const int hf = lane >> 4, lid = lane & 15;
  const int i0 = blockIdx.x * 16, nt = blockIdx.y, h = blockIdx.z;
  const float* arow = attn + (size_t)h * LEN * LEN + (size_t)(i0 + lid) * LEN + 2 * hf;
  int col;
  bool valid;
  const float* brow;
  if (nt == 0) { col = lid; valid = true;
                 brow = vsT + ((size_t)h * DSV + col) * LEN + 2 * hf; }
  else         { col = (nt - 1) * 16 + lid; valid = col < DPV3;
                 brow = vpT + ((size_t)h * DPV3 + (valid ? col : 0)) * LEN + 2 * hf; }
  v8f acc = {};
  for (int k = 0; k < LEN; k += 4) {
    v2f a = *(const v2f*)(arow + k);
    v2f b;
    if (valid) b = *(const v2f*)(brow + k);
    else       { b.x = 0.f; b.y = 0.f; }
    acc = wmma4(a, b, acc);
  }
  if (!valid) return;
  float* out = (nt == 0) ? (res_s + (size_t)h * LEN * DSV + col)
                         : (res_p + (size_t)h * LEN * DPV3 + col);
  const int os = (nt == 0) ? DSV : DPV3;
#pragma unroll
  for (int r = 0; r < 8; ++r) out[(size_t)(i0 + 8 * hf + r) * os] = acc[r];
}

// ---------------------------------------------------------------------------
// Kernel 7: res_pair[:,i,:] = attn[:,i,:] (12x512 padded to 16) @ pair[i] (512x128)
// B-panel (512 rows x 16 cols, row stride 128) TDM-staged into LDS (32 KB).
// ---------------------------------------------------------------------------
__global__ __launch_bounds__(32) void k_attn_pair(const float* __restrict__ attn,
                                                  const float* __restrict__ pair,
                                                  float* __restrict__ res_pair) {
  __shared__ float smem[LEN * 16];  // 32 KB B-panel, row-major [k][c]
  const int lane = threadIdx.x & 31;
  const int hf = lane >> 4, lid = lane & 15;
  const int n0 = blockIdx.x * 16, i = blockIdx.y;
  stage_tile(pair + (size_t)i * LEN * DPAIR + n0, smem, 16, LEN, DPAIR);
  const bool avalid = lid < NH;  // M rows are heads
  const float* arow = attn + (size_t)lid * LEN * LEN + (size_t)i * LEN + 2 * hf;
  v8f acc = {};
  for (int k = 0; k < LEN; k += 4) {
    v2f a;
    if (avalid) a = *(const v2f*)(arow + k);
    else        { a.x = 0.f; a.y = 0.f; }
    v2f b;
    b.x = smem[(k + 2 * hf) * 16 + lid];       // ds_load_b32
    b.y = smem[(k + 2 * hf + 1) * 16 + lid];
    acc = wmma4(a, b, acc);
  }
  const int d = n0 + lid;
#pragma unroll
  for (int r = 0; r < 8; ++r) {
    int h = 8 * hf + r;
    if (h < NH) res_pair[((size_t)h * LEN + i) * DPAIR + d] = acc[r];
  }
}

// ---------------------------------------------------------------------------
// Kernel 8: inverse rotation of res_p, norms, and feature concat F[i][2112].
// One thread per (h,i).
// ---------------------------------------------------------------------------
__global__ void k_finalize(const float* __restrict__ res_s, const float* __restrict__ res_p,
                           const float* __restrict__ res_pair, const float* __restrict__ rot,
                           const float* __restrict__ trans, float* __restrict__ F) {
  int idx = blockIdx.x * blockDim.x + threadIdx.x;
  if (idx >= NH * LEN) return;
  int h = idx / LEN, i = idx % LEN;
  float* fr = F + (size_t)i * DFEAT;
  const float* rs = res_s + ((size_t)h * LEN + i) * DSV;
#pragma unroll
  for (int d = 0; d < DSV; ++d) fr[h * DSV + d] = rs[d];
  const float* R = rot + (size_t)i * 9;
  const float* t = trans + (size_t)i * 3;
  const float* rp = res_p + ((size_t)h * LEN + i) * DPV3;
#pragma unroll
  for (int p = 0; p < 8; ++p) {
    float c0 = rp[p * 3 + 0] - t[0];
    float c1 = rp[p * 3 + 1] - t[1];
    float c2 = rp[p * 3 + 2] - t[2];
    // einsum 'bhlpc,bldc->bhlpd': out_d = sum_c v_c * R[d][c]
    float d0 = c0 * R[0] + c1 * R[1] + c2 * R[2];
    float d1 = c0 * R[3] + c1 * R[4] + c2 * R[5];
    float d2 = c0 * R[6] + c1 * R[7] + c2 * R[8];
    fr[192 + h * 24 + p * 3 + 0] = d0;
    fr[192 + h * 24 + p * 3 + 1] = d1;
    fr[192 + h * 24 + p * 3 + 2] = d2;
    fr[480 + h * 8 + p] = sqrtf(d0 * d0 + d1 * d1 + d2 * d2 + EPSV);
  }
  const float* rpr = res_pair + ((size_t)h * LEN + i) * DPAIR;
  for (int d = 0; d < DPAIR; ++d) fr[576 + h * DPAIR + d] = rpr[d];
}

// ---------------------------------------------------------------------------
// Kernel 9: out = F (512x2112) @ WoutT^T (+ b_out), transposed-B b64 loads.
// ---------------------------------------------------------------------------
__global__ __launch_bounds__(32) void k_outT(const float* __restrict__ F,
                                             const float* __restrict__ WT,
                                             const float* __restrict__ b,
                                             float* __restrict__ out) {
  const int lane = threadIdx.x & 31;
  const int hf = lane >> 4, lid = lane & 15;
  const int m0 = blockIdx.x * 16, n0 = blockIdx.y * 16;
  const float* xrow = F + (size_t)(m0 + lid) * DFEAT + 2 * hf;
  const float* brow = WT + (size_t)(n0 + lid) * DFEAT + 2 * hf;
  v8f acc = {};
  for (int k = 0; k < DFEAT; k += 4) {
    v2f a = *(const v2f*)(xrow + k);
    v2f b2 = *(const v2f*)(brow + k);
    acc = wmma4(a, b2, acc);
  }
  const float bb = b[n0 + lid];
  float* o = out + n0 + lid;
#pragma unroll
  for (int r = 0; r < 8; ++r) o[(size_t)(m0 + 8 * hf + r) * DMODEL] = acc[r] + bb;
}

// ---------------------------------------------------------------------------
// Launch
// ---------------------------------------------------------------------------
extern "C" void kernel_launch(void* const* d_in, const int* in_sizes, int n_in,
                              void* d_out, int out_size, void* d_ws, size_t ws_size,
                              hipStream_t stream) {
  const float* x     = (const float*)d_in[0];
  const float* pair  = (const float*)d_in[1];
  const float* rot   = (const float*)d_in[2];
  const float* trans = (const float*)d_in[3];
  const float* Wqs   = (const float*)d_in[4];
  const float* Wqp   = (const float*)d_in[5];
  const float* pwraw = (const float*)d_in[6];
  const float* Wpair = (const float*)d_in[7];
  const float* bpair = (const float*)d_in[8];
  const float* Wout  = (const float*)d_in[9];
  const float* bout  = (const float*)d_in[10];
  float* out = (float*)d_out;
  float* ws  = (float*)d_ws;

  // workspace offsets (floats); total ~10.9M floats = 43.4 MB
  constexpr size_t o_qkvs = 0;                                   // 512*576
  constexpr size_t o_qkvp = o_qkvs + (size_t)LEN * NSOUT;
  constexpr size_t o_qs   = o_qkvp + (size_t)LEN * NPOUT;
  constexpr size_t o_ks   = o_qs   + (size_t)NH * LEN * DSK;
  constexpr size_t o_vsT  = o_ks   + (size_t)NH * LEN * DSK;     // [h][d][i]
  constexpr size_t o_qp   = o_vsT  + (size_t)NH * DSV * LEN;
  constexpr size_t o_kp   = o_qp   + (size_t)NH * LEN * DPK3;
  constexpr size_t o_vpT  = o_kp   + (size_t)NH * LEN * DPK3;    // [h][c][i]
  constexpr size_t o_qp2  = o_vpT  + (size_t)NH * DPV3 * LEN;
  constexpr size_t o_kp2  = o_qp2  + (size_t)NH * LEN;
  constexpr size_t o_pb   = o_kp2  + (size_t)NH * LEN;           // NH*LEN*LEN
  constexpr size_t o_log  = o_pb   + (size_t)NH * LEN * LEN;     // NH*LEN*LEN
  constexpr size_t o_ress = o_log  + (size_t)NH * LEN * LEN;
  constexpr size_t o_resp = o_ress + (size_t)NH * LEN * DSV;
  constexpr size_t o_rpr  = o_resp + (size_t)NH * LEN * DPV3;
  constexpr size_t o_F    = o_rpr  + (size_t)NH * LEN * DPAIR;   // LEN*2112
  constexpr size_t o_WqsT = o_F    + (size_t)LEN * DFEAT;        // 576*384
  constexpr size_t o_WqpT = o_WqsT + (size_t)NSOUT * DMODEL;
  constexpr size_t o_WoT  = o_WqpT + (size_t)NPOUT * DMODEL;     // 384*2112

  dim3 wave(32);
  // 0) weight transposes (B-operands become contiguous b64 loads)
  k_transpose<<<(DMODEL * NSOUT + 255) / 256, 256, 0, stream>>>(Wqs, ws + o_WqsT, DMODEL, NSOUT);
  k_transpose<<<(DMODEL * NPOUT + 255) / 256, 256, 0, stream>>>(Wqp, ws + o_WqpT, DMODEL, NPOUT);
  k_transpose<<<(DFEAT * DMODEL + 255) / 256, 256, 0, stream>>>(Wout, ws + o_WoT, DFEAT, DMODEL);
  // 1) qkv projections (fp32 WMMA GEMMs)
  k_gemmT<<<dim3(LEN / 16, NSOUT / 16), wave, 0, stream>>>(x, ws + o_WqsT, ws + o_qkvs, DMODEL, NSOUT);
  k_gemmT<<<dim3(LEN / 16, NPOUT / 16), wave, 0, stream>>>(x, ws + o_WqpT, ws + o_qkvp, DMODEL, NPOUT);
  // 2) head scatter + point rotation + norms
  k_prep<<<(NH * LEN + 255) / 256, 256, 0, stream>>>(
      ws + o_qkvs, ws + o_qkvp, rot, trans, ws + o_qs, ws + o_ks, ws + o_vsT,
      ws + o_qp, ws + o_kp, ws + o_vpT, ws + o_qp2, ws + o_kp2);
  // 3) pair bias (TDM-streams 134MB pairwise through LDS)
  k_pairbias<<<dim3(LEN / 16, LEN), wave, 0, stream>>>(pair, Wpair, bpair, ws + o_pb);
  // 4) logits (scalar QK + point cross-term via WMMA)
  k_logits<<<dim3(LEN / 16, LEN / 16, NH), wave, 0, stream>>>(
      ws + o_qs, ws + o_ks, ws + o_qp, ws + o_kp, ws + o_qp2, ws + o_kp2,
      pwraw, ws + o_pb, ws + o_log);
  // 5) softmax (in place -> attn)
  k_softmax<<<(NH * LEN) / 8, 256, 0, stream>>>(ws + o_log);
  // 6) attn @ [v_s | v_p]
  k_attn_v<<<dim3(LEN / 16, 3, NH), wave, 0, stream>>>(
      ws + o_log, ws + o_vsT, ws + o_vpT, ws + o_ress, ws + o_resp);
  // 7) attn @ pairwise (per-i GEMM, B-panel via TDM, pairwise hits L2)
  k_attn_pair<<<dim3(DPAIR / 16, LEN), wave, 0, stream>>>(ws + o_log, pair, ws + o_rpr);
  // 8) inverse rotation, norms, concat features
  k_finalize<<<(NH * LEN + 255) / 256, 256, 0, stream>>>(
      ws + o_ress, ws + o_resp, ws + o_rpr, rot, trans, ws + o_F);
  // 9) output projection + bias
  k_outT<<<dim3(LEN / 16, DMODEL / 16), wave, 0, stream>>>(ws + o_F, ws + o_WoT, bout, out);
}